// HCNN_KNOWN_U_26156350832749
// MI455X (gfx1250) — compile-verified
//
#include <hip/hip_runtime.h>
#include <math.h>

// ---------------------------------------------------------------------------
// HCNN recurrence: 80 steps of  x = tanh(splice(state,Y,U));  state = x @ A
//   N_STATE=2048, N_U=64, N_Y=128, PAST=64, FORE=16, BATCH=256
// GEMM per step: M=256, K=2112, N=2048, exact fp32 via V_WMMA_F32_16X16X4_F32.
// Split-K=2 for occupancy; split-K reduction fused into the next prep kernel.
// K-loop unrolled 2x so the second group's loads overlap the first's WMMAs.
// ---------------------------------------------------------------------------

#define N_STATE 2048
#define N_U     64
#define N_Y     128
#define PAST    64
#define FORE    16
#define BATCH   256
#define K_DIM   (N_STATE + N_U)   // 2112
#define K_HALF  (K_DIM / 2)       // 1056 (divisible by 8)

typedef float v2f __attribute__((ext_vector_type(2)));
typedef float v8f __attribute__((ext_vector_type(8)));

// ---------------------------------------------------------------------------
// prep: state = P0 + P1 (split-K partials), then
//   past: err = state[:,:128] - y -> out; x = tanh([y | state[128:] | u])
//   fore: out = state[:,:128];           x = tanh([state | u])
// Grid: (ceil(K_DIM/256), BATCH); one thread per x element, no int division.
// S1 == nullptr on step 0 (init state, row-stride 0 broadcast).
// ---------------------------------------------------------------------------
__global__ __launch_bounds__(256) void hcnn_prep(
    const float* __restrict__ S0, const float* __restrict__ S1, int s_row_stride,
    const float* __restrict__ u,                       // (BATCH, N_U)
    const float* __restrict__ y,                       // (BATCH, N_Y) or null
    float* __restrict__ X,                             // (BATCH, K_DIM)
    float* __restrict__ out_t,                         // (BATCH, N_Y)
    int is_past)
{
    int c = blockIdx.x * 256 + threadIdx.x;            // column in x
    int b = blockIdx.y;                                // batch row
    if (c >= K_DIM) return;

    float v;
    if (c < N_STATE) {
        float s = S0[b * s_row_stride + c];
        if (S1) s += S1[b * s_row_stride + c];         // fused split-K reduce
        if (c < N_Y) {
            if (is_past) {
                float yv = y[b * N_Y + c];
                out_t[b * N_Y + c] = s - yv;           // err
                v = yv;                                // splice with y
            } else {
                out_t[b * N_Y + c] = s;                // expectation
                v = s;
            }
        } else {
            v = s;
        }
    } else {
        v = u[b * N_U + (c - N_STATE)];
    }
    X[b * K_DIM + c] = tanhf(v);
}

// ---------------------------------------------------------------------------
// GEMM: Ppart[z] = X[:, zK:(z+1)K] @ A[zK:(z+1)K, :], fp32 WMMA 16x16x4.
// Block = 128 threads (4 waves). Each wave: 32(M) x 64(N) = 8 output tiles,
// so the 4 B-fragments per K-step feed 2 M-tiles (1.25 loads/WMMA).
// grid = (BATCH/32, N_STATE/256, 2 split-K) -> 512 waves.
//
// f32 WMMA fragment layouts (ISA 7.12.2):
//   A (16x4, MxK):  lanes 0-15: v0=K0,v1=K1 for M=lane; lanes 16-31: K2/K3.
//   B (4x16, KxN):  lanes 0-15 hold low K pair, lanes 16-31 high pair; N=lane%16.
//   C/D (16x16):    vgpr r: lanes 0-15 -> M=r, lanes 16-31 -> M=r+8; N=lane%16.
// ---------------------------------------------------------------------------
__global__ __launch_bounds__(128) void hcnn_gemm(
    const float* __restrict__ X,    // (BATCH, K_DIM)
    const float* __restrict__ A,    // (K_DIM, N_STATE)
    float* __restrict__ Ppart)      // (2, BATCH, N_STATE) split-K partials
{
    const int lane  = threadIdx.x & 31;
    const int wave  = threadIdx.x >> 5;
    const int mrow  = lane & 15;
    const int khalf = lane >> 4;        // 0: K0/K1, 1: K2/K3

    const int kz   = blockIdx.z;        // split-K index
    const int kbeg = kz * K_HALF;
    const int kend = kbeg + K_HALF;
    float* P = Ppart + (size_t)kz * BATCH * N_STATE;

    const int mbase = blockIdx.x * 32;
    const int nbase = blockIdx.y * 256 + wave * 64;

    const float* xrow0 = X + (size_t)(mbase + mrow) * K_DIM + 2 * khalf;
    const float* xrow1 = xrow0 + (size_t)16 * K_DIM;
    const float* acol  = A + (size_t)(2 * khalf) * N_STATE + nbase + mrow;

    v8f acc[2][4] = {};

    // 2x unroll: the second K-group's loads can issue above the first
    // group's WMMAs, hiding L2 latency inside a single wave.
    #pragma unroll 2
    for (int k = kbeg; k < kend; k += 4) {
        float2 a0 = *(const float2*)(xrow0 + k);     // A frags (8B aligned)
        float2 a1 = *(const float2*)(xrow1 + k);
        v2f av0; av0.x = a0.x; av0.y = a0.y;
        v2f av1; av1.x = a1.x; av1.y = a1.y;

        const float* ap = acol + (size_t)k * N_STATE;
        v2f bv[4];
        #pragma unroll
        for (int j = 0; j < 4; ++j) {
            bv[j].x = ap[j * 16];
            bv[j].y = ap[j * 16 + N_STATE];
        }

        #pragma unroll
        for (int j = 0; j < 4; ++j) {
            acc[0][j] = __builtin_amdgcn_wmma_f32_16x16x4_f32(
                false, av0, false, bv[j], (short)0, acc[0][j], false, false);
            acc[1][j] = __builtin_amdgcn_wmma_f32_16x16x4_f32(
                false, av1, false, bv[j], (short)0, acc[1][j], false, false);
        }
    }

    #pragma unroll
    for (int i = 0; i < 2; ++i) {
        const int mb = mbase + i * 16 + 8 * khalf;
        #pragma unroll
        for (int r = 0; r < 8; ++r) {
            float* srow = P + (size_t)(mb + r) * N_STATE + nbase + mrow;
            srow[0]  = acc[i][0][r];
            srow[16] = acc[i][1][r];
            srow[32] = acc[i][2][r];
            srow[48] = acc[i][3][r];
        }
    }
}

// ---------------------------------------------------------------------------
// kernel_launch: 80 x (prep [+ fused split-K reduce]) + 79 x GEMM.
// d_ws layout: [ P0 | P1 : each 256*2048 f32 ][ X : 256*2112 f32 ]  (~6.4 MB)
// ---------------------------------------------------------------------------
extern "C" void kernel_launch(void* const* d_in, const int* in_sizes, int n_in,
                              void* d_out, int out_size, void* d_ws, size_t ws_size,
                              hipStream_t stream) {
    (void)in_sizes; (void)n_in; (void)out_size; (void)ws_size;

    const float* U    = (const float*)d_in[0];   // (80, 256, 64)
    const float* Y    = (const float*)d_in[1];   // (64, 256, 128)
    const float* A    = (const float*)d_in[2];   // (2112, 2048)
    const float* init = (const float*)d_in[3];   // (1, 2048)
    float* out = (float*)d_out;                  // (80, 256, 128)

    float* P0   = (float*)d_ws;                          // BATCH * N_STATE
    float* P1   = P0 + (size_t)BATCH * N_STATE;          // BATCH * N_STATE
    float* Xbuf = P1 + (size_t)BATCH * N_STATE;          // BATCH * K_DIM

    const float* cur0 = init;                    // step 0: broadcast init row
    const float* cur1 = nullptr;
    int stride = 0;

    const dim3 prep_grid((K_DIM + 255) / 256, BATCH);
    const dim3 gemm_grid(BATCH / 32, N_STATE / 256, 2);

    for (int t = 0; t < PAST + FORE; ++t) {
        const float* u  = U + (size_t)t * BATCH * N_U;
        const float* y  = (t < PAST) ? (Y + (size_t)t * BATCH * N_Y) : nullptr;
        float* out_t    = out + (size_t)t * BATCH * N_Y;

        hcnn_prep<<<prep_grid, 256, 0, stream>>>(cur0, cur1, stride, u, y,
                                                 Xbuf, out_t, (t < PAST) ? 1 : 0);

        if (t < PAST + FORE - 1) {
            hcnn_gemm<<<gemm_grid, 128, 0, stream>>>(Xbuf, A, P0);
            cur0 = P0; cur1 = P1;
            stride = N_STATE;
        }
    }
}